// CodedNet_55533927137718
// MI455X (gfx1250) — compile-verified
//
#include <hip/hip_runtime.h>
#include <stdint.h>

// out[b,i,j] = sum_c sign(w)[(i-c) mod 32, j mod 32] * x[b,i,j,c]
// (the forward and backward per-channel rolls cancel on x; only the mask rotates)

namespace {
constexpr int kC          = 31;
constexpr int kPixels     = 256 * 64 * 64;          // B*P*P = 1,048,576
constexpr int kTilePixels = 128;
constexpr int kTileDwords = kTilePixels * kC;       // 3968 dwords = 15,872 B (16B aligned)
constexpr int kTileF4     = kTileDwords / 4;        // 992 float4s
constexpr int kThreads    = 128;                    // 4 waves (wave32)
constexpr int kF4PerThread = (kTileF4 + kThreads - 1) / kThreads;  // 8 (uniform per wave)
constexpr int kNumTiles   = kPixels / kTilePixels;  // 8192
constexpr int kTilesPerBlock = 8;
constexpr int kBlocks     = kNumTiles / kTilesPerBlock;  // 1024
}

// CDNA5 async global->LDS copy (ASYNCcnt-tracked). GV mode: 64-bit vaddr, saddr=off.
__device__ __forceinline__ void async_b128_to_lds(unsigned lds_addr, const void* gaddr) {
  asm volatile("global_load_async_to_lds_b128 %0, %1, off"
               :: "v"(lds_addr), "v"(gaddr)
               : "memory");
}

__global__ __launch_bounds__(kThreads)
void codednet_rollmask_kernel(const float* __restrict__ x,
                              const float* __restrict__ w,
                              float* __restrict__ out) {
  __shared__ float s_sign[1024];                 // sign(w) as 32x32 table, 4 KB
  __shared__ float s_xtile[2][kTileDwords];      // double-buffered pixel tiles, 2x15.5 KB

  const int tid = threadIdx.x;

  // Stage sign(w) into LDS (handles sign(0)=0 exactly).
  #pragma unroll
  for (int u = 0; u < 1024 / kThreads; ++u) {
    const int k = tid + kThreads * u;
    const float wv = w[k];
    s_sign[k] = (wv > 0.0f) ? 1.0f : ((wv < 0.0f) ? -1.0f : 0.0f);
  }

  const int tile0 = blockIdx.x * kTilesPerBlock;

  auto issue_tile = [&](int rel, int buf) {
    const float* gbase = x + (size_t)(tile0 + rel) * kTileDwords;
    const unsigned lbase = (unsigned)(uintptr_t)(&s_xtile[buf][0]);
    #pragma unroll
    for (int u = 0; u < kF4PerThread; ++u) {
      int idx = tid + kThreads * u;
      if (idx >= kTileF4) idx -= kTileF4;  // benign duplicate copy: keeps per-wave ASYNCcnt uniform
      async_b128_to_lds(lbase + (unsigned)idx * 16u, gbase + (size_t)idx * 4);
    }
  };

  issue_tile(0, 0);  // prime the pipeline

  for (int t = 0; t < kTilesPerBlock; ++t) {
    const bool more = (t + 1 < kTilesPerBlock);
    if (more) {
      issue_tile(t + 1, (t + 1) & 1);
      // Async loads complete in order within a wave: outstanding <= 8 (the
      // just-issued tile t+1 batch) implies all of tile t's data is in LDS.
      asm volatile("s_wait_asynccnt 8" ::: "memory");
    } else {
      asm volatile("s_wait_asynccnt 0" ::: "memory");
    }
    __syncthreads();  // make every wave's async-landed LDS data (and s_sign) visible

    // Compute: one pixel per thread, conflict-free LDS reads (stride-31 is odd).
    const float* xt = &s_xtile[t & 1][0];
    const size_t p  = (size_t)(tile0 + t) * kTilePixels + (size_t)tid;
    const int r0 = (int)((p >> 6) & 31);  // i mod 32
    const int jj = (int)(p & 31);         // j mod 32
    float acc = 0.0f;
    #pragma unroll
    for (int c = 0; c < kC; ++c) {
      const float xv = xt[tid * kC + c];
      const float sv = s_sign[(((r0 - c) & 31) << 5) | jj];
      acc = __builtin_fmaf(xv, sv, acc);
    }
    out[p] = acc;  // coalesced: 128 consecutive dwords per tile

    __syncthreads();  // protect buffer (t&1) before tile t+2's async writes reuse it
  }
}

extern "C" void kernel_launch(void* const* d_in, const int* in_sizes, int n_in,
                              void* d_out, int out_size, void* d_ws, size_t ws_size,
                              hipStream_t stream) {
  (void)in_sizes; (void)n_in; (void)out_size; (void)d_ws; (void)ws_size;
  const float* x = (const float*)d_in[0];
  const float* w = (const float*)d_in[1];
  float* out    = (float*)d_out;
  codednet_rollmask_kernel<<<dim3(kBlocks), dim3(kThreads), 0, stream>>>(x, w, out);
}